// Estor_concat_45595372814584
// MI455X (gfx1250) — compile-verified
//
#include <hip/hip_runtime.h>

typedef __attribute__((ext_vector_type(2))) float v2f;
typedef __attribute__((ext_vector_type(8))) float v8f;
typedef int v4i __attribute__((vector_size(4 * sizeof(int))));
typedef __attribute__((address_space(1))) v4i gv4i;   // global 4xi32 vector
typedef __attribute__((address_space(3))) v4i lv4i;   // LDS    4xi32 vector

#define Tq    16
#define Bq    8
#define Sq    512
#define Hq    768
#define H2q   384
#define NEWH  1152
#define NLAB  33
#define NSPANS 4096
#define PPAD  36   // padded panel row stride (floats): 16B-aligned + conflict-free

#if defined(__has_builtin)
#if __has_builtin(__builtin_amdgcn_global_load_async_to_lds_b128)
#define HAVE_ASYNC_LDS 1
#endif
#endif
#ifndef HAVE_ASYNC_LDS
#define HAVE_ASYNC_LDS 0
#endif

static __device__ inline void wait_asynccnt0() {
#if defined(__has_builtin) && __has_builtin(__builtin_amdgcn_s_wait_asynccnt)
  __builtin_amdgcn_s_wait_asynccnt(0);
#else
  asm volatile("s_wait_asynccnt 0x0" ::: "memory");
#endif
}

// D = A(16x4,f32) * B(4x16,f32) + C   -- CDNA5 V_WMMA_F32_16X16X4_F32
static __device__ inline v8f wmma4(v2f a, v2f b, v8f c) {
  return __builtin_amdgcn_wmma_f32_16x16x4_f32(false, a, false, b, (short)0, c,
                                               false, false);
}

// -------------------------------------------------------------------------
__global__ void k_zero(float* __restrict__ p, int n) {
  int i = blockIdx.x * blockDim.x + threadIdx.x;
  if (i < n) p[i] = 0.0f;
}

// span scatter: counts[t,b,s] += 1 for s in [start,end)
__global__ void k_spans(const int* __restrict__ sb, const int* __restrict__ st,
                        const int* __restrict__ ss, const int* __restrict__ se,
                        float* __restrict__ counts) {
  int i = blockIdx.x * blockDim.x + threadIdx.x;
  if (i >= NSPANS) return;
  int b = sb[i], t = st[i], s0 = ss[i], s1 = se[i];
  float* base = counts + (size_t)t * (Bq * Sq) + (size_t)b * Sq;
  for (int s = s0; s < s1 && s < Sq; ++s) atomicAdd(base + s, 1.0f);
}

// -------------------------------------------------------------------------
// D[t, h0+n] = sum_k A[t,k] * Bm[h0+n, k] + bias[h0+n]; one wave per 16-tile
__global__ void __launch_bounds__(32)
k_gemm16(const float* __restrict__ A, const float* __restrict__ Bm,
         const float* __restrict__ bias, float* __restrict__ D) {
  const int h0   = blockIdx.x * 16;
  const int lane = threadIdx.x;
  const int half = (lane >= 16) ? 2 : 0;
  const int col  = lane & 15;
  v8f acc = {};
  for (int k0 = 0; k0 < Hq; k0 += 4) {
    v2f a, b;
    a.x = A[col * Hq + k0 + half];
    a.y = A[col * Hq + k0 + half + 1];
    b.x = Bm[(size_t)(h0 + col) * Hq + k0 + half];
    b.y = Bm[(size_t)(h0 + col) * Hq + k0 + half + 1];
    acc = wmma4(a, b, acc);
  }
  const float bb = bias[h0 + col];
#pragma unroll
  for (int r = 0; r < 8; ++r) {
    int m = r + ((lane < 16) ? 0 : 8);
    D[m * Hq + h0 + col] = acc[r] + bb;
  }
}

// -------------------------------------------------------------------------
// M[t,o] = sum_h v_tag[t,h] * ff1_w[o, t*H + h] : single-use stream of 37.7 MB
__global__ void __launch_bounds__(256)
k_build_M(const float* __restrict__ ff1_w, const float* __restrict__ v_tag,
          float* __restrict__ M) {
  const int wave = threadIdx.x >> 5;
  const int lane = threadIdx.x & 31;
  const int row  = blockIdx.x * 8 + wave;
  const int t = row / Hq;
  const int o = row - t * Hq;
  const float* __restrict__ wrow = ff1_w + (size_t)o * (Tq * Hq) + (size_t)t * Hq;
  const float* __restrict__ vrow = v_tag + (size_t)t * Hq;
  float s = 0.0f;
  for (int h = lane; h < Hq; h += 32) {
    __builtin_prefetch(wrow + h + 256, 0, 0);  // global_prefetch_b8
    s += wrow[h] * vrow[h];
  }
#pragma unroll
  for (int off = 16; off > 0; off >>= 1) s += __shfl_xor(s, off, 32);
  if (lane == 0) M[row] = s;
}

// -------------------------------------------------------------------------
// fused per-position pipeline: 16 positions per block.
//  stage1: h1(16x768) = relu(counts(16x16) @ M + ff1_b)          [WMMA K=16]
//  stage2: h2(16x384) = h1 @ ff2_w^T + ff2_b                     [WMMA K=768,
//          B K-panels streamed to LDS via GLOBAL_LOAD_ASYNC_TO_LDS_B128]
//  stage3: layernorm(concat(word_emb, h2)) @ lin_w^T + lin_b
__global__ void __launch_bounds__(128)
k_fused(const float* __restrict__ counts, const float* __restrict__ M,
        const float* __restrict__ ff1_b, const float* __restrict__ ff2_w,
        const float* __restrict__ ff2_b, const float* __restrict__ word_emb,
        const float* __restrict__ ln_g, const float* __restrict__ ln_b,
        const float* __restrict__ lin_w, const float* __restrict__ lin_b,
        float* __restrict__ out) {
  extern __shared__ float smem[];
  float* bufA  = smem;                           // 16*1152: h1, later normalized x
  float* h2s   = smem + 16 * NEWH;               // 16*384
  float* panel = h2s + 16 * H2q;                 // 384*PPAD: ff2_w K-panel
  __shared__ float cnt[16 * 16];

  const int tid  = threadIdx.x;
  const int lane = tid & 31;
  const int wave = tid >> 5;
  const int half = (lane >= 16) ? 2 : 0;
  const int col  = lane & 15;
  const int blk  = blockIdx.x;                   // 256 = 8 b * 32 s-tiles
  const int b    = blk >> 5;
  const int s0   = (blk & 31) << 4;

  for (int i = tid; i < 256; i += 128) {
    int p = i >> 4, t = i & 15;
    cnt[p * 16 + t] = counts[(size_t)t * (Bq * Sq) + (size_t)b * Sq + s0 + p];
  }
  __syncthreads();

  // ---- stage 1: h1 = relu(cnt @ M + ff1_b), 48 o-tiles over 4 waves
  for (int ot = wave; ot < 48; ot += 4) {
    const int o0 = ot * 16;
    v8f acc = {};
#pragma unroll
    for (int k0 = 0; k0 < 16; k0 += 4) {
      v2f a, bm;
      a.x  = cnt[col * 16 + k0 + half];
      a.y  = cnt[col * 16 + k0 + half + 1];
      bm.x = M[(size_t)(k0 + half) * Hq + o0 + col];
      bm.y = M[(size_t)(k0 + half + 1) * Hq + o0 + col];
      acc = wmma4(a, bm, acc);
    }
    const float bb = ff1_b[o0 + col];
#pragma unroll
    for (int r = 0; r < 8; ++r) {
      int m = r + ((lane < 16) ? 0 : 8);
      float h = acc[r] + bb;
      bufA[m * Hq + o0 + col] = (h > 0.0f) ? h : 0.0f;
    }
  }

  // ---- stage 2: h2 = h1 @ ff2_w^T; K swept in 24 panels of 32, staged in LDS
  v8f acc2[6];
#pragma unroll
  for (int q = 0; q < 6; ++q) acc2[q] = (v8f){};

  for (int kp = 0; kp < Hq; kp += 32) {
    __syncthreads();  // previous panel consumed (also orders stage-1 bufA)
    // cooperative copy: ff2_w[:, kp:kp+32] -> panel[j*PPAD + kl], 128b/lane
    for (int i = tid; i < H2q * 8; i += 128) {
      int j  = i >> 3;
      int k4 = (i & 7) << 2;
#if HAVE_ASYNC_LDS
      __builtin_amdgcn_global_load_async_to_lds_b128(
          (gv4i*)(ff2_w + (size_t)j * Hq + kp + k4),
          (lv4i*)(panel + j * PPAD + k4), 0, 0);
#else
      const float4 vv = *(const float4*)(ff2_w + (size_t)j * Hq + kp + k4);
      *(float4*)(panel + j * PPAD + k4) = vv;
#endif
    }
#if HAVE_ASYNC_LDS
    wait_asynccnt0();
#endif
    __syncthreads();  // panel visible to all waves

#pragma unroll
    for (int q = 0; q < 6; ++q) {
      const int j0 = (wave + 4 * q) * 16;
#pragma unroll
      for (int kl = 0; kl < 32; kl += 4) {
        v2f a, bm;
        a.x  = bufA[col * Hq + kp + kl + half];
        a.y  = bufA[col * Hq + kp + kl + half + 1];
        bm.x = panel[(j0 + col) * PPAD + kl + half];
        bm.y = panel[(j0 + col) * PPAD + kl + half + 1];
        acc2[q] = wmma4(a, bm, acc2[q]);
      }
    }
  }

#pragma unroll
  for (int q = 0; q < 6; ++q) {
    const int j0 = (wave + 4 * q) * 16;
    const float bb = ff2_b[j0 + col];
#pragma unroll
    for (int r = 0; r < 8; ++r) {
      int m = r + ((lane < 16) ? 0 : 8);
      h2s[m * H2q + j0 + col] = acc2[q][r] + bb;
    }
  }
  __syncthreads();

  // ---- stage 3: layernorm + linear head; 8 threads per position
  const int p  = tid >> 3;
  const int l8 = tid & 7;
  const float* __restrict__ wrow = word_emb + ((size_t)b * Sq + s0 + p) * Hq;
  float s1 = 0.0f, s2 = 0.0f;
  for (int j = l8; j < NEWH; j += 8) {
    float x = (j < Hq) ? wrow[j] : h2s[p * H2q + (j - Hq)];
    s1 += x;
    s2 += x * x;
  }
#pragma unroll
  for (int off = 4; off > 0; off >>= 1) {
    s1 += __shfl_xor(s1, off, 8);
    s2 += __shfl_xor(s2, off, 8);
  }
  const float mu   = s1 * (1.0f / NEWH);
  const float var  = s2 * (1.0f / NEWH) - mu * mu;
  const float rstd = rsqrtf(var + 1e-12f);
  for (int j = l8; j < NEWH; j += 8) {
    float x = (j < Hq) ? wrow[j] : h2s[p * H2q + (j - Hq)];
    bufA[p * NEWH + j] = (x - mu) * rstd * ln_g[j] + ln_b[j];  // h1 region dead
  }
  __syncthreads();

  const float* __restrict__ xn = bufA + p * NEWH;
  float* __restrict__ orow = out + ((size_t)b * Sq + s0 + p) * NLAB;
  for (int l = l8; l < NLAB; l += 8) {
    const float* __restrict__ wl = lin_w + (size_t)l * NEWH;
    float acc = lin_b[l];
    for (int j = 0; j < NEWH; ++j) acc += xn[j] * wl[j];
    orow[l] = acc;
  }
}

// -------------------------------------------------------------------------
extern "C" void kernel_launch(void* const* d_in, const int* in_sizes, int n_in,
                              void* d_out, int out_size, void* d_ws, size_t ws_size,
                              hipStream_t stream) {
  const float* word_emb   = (const float*)d_in[0];
  const float* tag_emb    = (const float*)d_in[1];
  const float* in_proj_w  = (const float*)d_in[2];
  const float* in_proj_b  = (const float*)d_in[3];
  const float* out_proj_w = (const float*)d_in[4];
  const float* out_proj_b = (const float*)d_in[5];
  const float* ff1_w      = (const float*)d_in[6];
  const float* ff1_b      = (const float*)d_in[7];
  const float* ff2_w      = (const float*)d_in[8];
  const float* ff2_b      = (const float*)d_in[9];
  const float* ln_g       = (const float*)d_in[10];
  const float* ln_b       = (const float*)d_in[11];
  const float* lin_w      = (const float*)d_in[12];
  const float* lin_b      = (const float*)d_in[13];
  const int* span_batch   = (const int*)d_in[14];
  const int* span_tag     = (const int*)d_in[15];
  const int* span_start   = (const int*)d_in[16];
  const int* span_end     = (const int*)d_in[17];

  float* ws     = (float*)d_ws;
  float* v      = ws;                 // 16*768
  float* v_tag  = ws + 12288;         // 16*768
  float* Mbuf   = ws + 24576;         // 16*768
  float* counts = ws + 36864;         // 16*8*512

  k_zero<<<256, 256, 0, stream>>>(counts, Tq * Bq * Sq);
  k_spans<<<16, 256, 0, stream>>>(span_batch, span_tag, span_start, span_end, counts);
  k_gemm16<<<48, 32, 0, stream>>>(tag_emb, in_proj_w + 2 * Hq * Hq,
                                  in_proj_b + 2 * Hq, v);
  k_gemm16<<<48, 32, 0, stream>>>(v, out_proj_w, out_proj_b, v_tag);
  k_build_M<<<1536, 256, 0, stream>>>(ff1_w, v_tag, Mbuf);

  const size_t shmem =
      (size_t)(16 * NEWH + 16 * H2q + H2q * PPAD) * sizeof(float);  // 153600 B
  k_fused<<<256, 128, shmem, stream>>>(counts, Mbuf, ff1_b, ff2_w, ff2_b, word_emb,
                                       ln_g, ln_b, lin_w, lin_b, (float*)d_out);
}